// SparseNeighbourhoodSelfAttn_32444182954023
// MI455X (gfx1250) — compile-verified
//
#include <hip/hip_runtime.h>
#include <hip/hip_bf16.h>
#include <math.h>

typedef __attribute__((ext_vector_type(2))) float v2f;
typedef __attribute__((ext_vector_type(8))) float v8f;

#define DMODEL 128
#define NHEADS 8
#define DK 16

// ---- order-preserving float <-> uint encoding for atomicMax on floats ----
__device__ __forceinline__ unsigned enc_ordered(float f) {
  int b = __float_as_int(f);
  return (b < 0) ? ~(unsigned)b : ((unsigned)b | 0x80000000u);
}
__device__ __forceinline__ float dec_ordered(unsigned u) {
  int b = (u & 0x80000000u) ? (int)(u & 0x7FFFFFFFu) : ~(int)u;
  return __int_as_float(b);
}

// ---- WMMA f32 GEMM: C[M x Ncols] = A[M x 128] * W[Ncols x 128]^T ----
// One wave (32 lanes) produces one 16x16 output tile via 32x V_WMMA_F32_16X16X4_F32.
// A-layout (16x4): lane&15 = M row, lane>>4 selects K pair {0,1} vs {2,3}, 2 VGPRs stripe K.
// B-layout (4x16): lane&15 = N col, lane>>4 selects K pair, 2 VGPRs stripe K.
// C/D layout: VGPR r holds rows (r, r+8), col = lane&15 (+lane>>4 selects +8 row).
template <int EPILOGUE>  // 0 = plain store, 1 = sigmoid(x + bias[col])
__global__ void wmma_gemm_xt(const float* __restrict__ A,
                             const float* __restrict__ W,
                             const float* __restrict__ bias,
                             float* __restrict__ C,
                             int M, int Ncols) {
  const int K = DMODEL;
  int wave  = (int)((blockIdx.x * blockDim.x + threadIdx.x) >> 5);
  int lane  = threadIdx.x & 31;
  int mtiles = (M + 15) >> 4;
  int ntiles = Ncols >> 4;
  int mt = wave / ntiles;
  int nt = wave - mt * ntiles;
  if (mt >= mtiles) return;

  int arow = mt * 16 + (lane & 15);
  if (arow >= M) arow = M - 1;                 // clamp (stores are guarded)
  int wrow = nt * 16 + (lane & 15);
  int kk   = (lane >> 4) * 2;                  // 0 or 2
  const float* Ap = A + (size_t)arow * K + kk;
  const float* Wp = W + (size_t)wrow * K + kk;

  v8f c = {};
#pragma unroll
  for (int k0 = 0; k0 < K; k0 += 4) {
    v2f a = *(const v2f*)(Ap + k0);
    v2f b = *(const v2f*)(Wp + k0);
    // 8 args: (neg_a, A, neg_b, B, c_mod, C, reuse_a, reuse_b)
    c = __builtin_amdgcn_wmma_f32_16x16x4_f32(false, a, false, b, (short)0, c,
                                              false, false);
  }

  int crow = mt * 16 + (lane >> 4) * 8;        // base row for this lane's 8 values
  int ccol = nt * 16 + (lane & 15);
  float bcol = (EPILOGUE == 1) ? bias[ccol] : 0.0f;
#pragma unroll
  for (int r = 0; r < 8; ++r) {
    int row = crow + r;
    if (row < M) {
      float v = c[r];
      if (EPILOGUE == 1) v = 1.0f / (1.0f + __expf(-(v + bcol)));
      C[(size_t)row * Ncols + ccol] = v;
    }
  }
}

// ---- per-(node,head) attention bias: bias[n,h] = X[n,:] . w_bias[h,:] ----
__global__ void bias_kernel(const float* __restrict__ X,
                            const float* __restrict__ w_bias,
                            float* __restrict__ bias, int M) {
  int i = blockIdx.x * blockDim.x + threadIdx.x;
  if (i >= M * NHEADS) return;
  int node = i >> 3, h = i & 7;
  const float4* xr = (const float4*)(X + (size_t)node * DMODEL);
  const float4* wr = (const float4*)(w_bias + (size_t)h * DMODEL);
  float acc = 0.f;
#pragma unroll
  for (int k = 0; k < DMODEL / 4; ++k) {
    float4 a = xr[k], b = wr[k];
    acc += a.x * b.x + a.y * b.y + a.z * b.z + a.w * b.w;
  }
  bias[i] = acc;
}

// ---- edge scores: s[e,h] = 0.25 * Q[src,h].K[dst,h] + bias[src,h]; segment max ----
__global__ void edge_scores(const float* __restrict__ qkv,
                            const float* __restrict__ bias,
                            const int* __restrict__ src,
                            const int* __restrict__ dst,
                            float* __restrict__ scores,
                            unsigned* __restrict__ mxbits, int E) {
  int i = blockIdx.x * blockDim.x + threadIdx.x;
  if (i >= E * NHEADS) return;
  int e = i >> 3, h = i & 7;
  int s = src[e], d = dst[e];
  const float4* q = (const float4*)(qkv + (size_t)s * (3 * DMODEL) + h * DK);
  const float4* k = (const float4*)(qkv + (size_t)d * (3 * DMODEL) + DMODEL + h * DK);
  float acc = 0.f;
#pragma unroll
  for (int j = 0; j < DK / 4; ++j) {
    float4 a = q[j], b = k[j];
    acc += a.x * b.x + a.y * b.y + a.z * b.z + a.w * b.w;
  }
  float sc = acc * 0.25f + bias[s * NHEADS + h];   // 1/sqrt(16) = 0.25
  scores[i] = sc;
  atomicMax(mxbits + s * NHEADS + h, enc_ordered(sc));
}

// ---- exp, segment sum, and unnormalized V aggregation ----
__global__ void edge_aggregate(const float* __restrict__ qkv,
                               const float* __restrict__ scores,
                               const unsigned* __restrict__ mxbits,
                               const int* __restrict__ src,
                               const int* __restrict__ dst,
                               float* __restrict__ sumexp,
                               float* __restrict__ accum, int E) {
  int i = blockIdx.x * blockDim.x + threadIdx.x;
  if (i >= E * NHEADS) return;
  int e = i >> 3, h = i & 7;
  int s = src[e], d = dst[e];
  float mx = dec_ordered(mxbits[s * NHEADS + h]);
  float ex = __expf(scores[i] - mx);
  atomicAdd(sumexp + s * NHEADS + h, ex);
  const float4* v = (const float4*)(qkv + (size_t)d * (3 * DMODEL) + 2 * DMODEL + h * DK);
  float* out = accum + (size_t)s * DMODEL + h * DK;
#pragma unroll
  for (int j = 0; j < DK / 4; ++j) {
    float4 vv = v[j];
    atomicAdd(out + j * 4 + 0, ex * vv.x);
    atomicAdd(out + j * 4 + 1, ex * vv.y);
    atomicAdd(out + j * 4 + 2, ex * vv.z);
    atomicAdd(out + j * 4 + 3, ex * vv.w);
  }
}

// ---- normalize by segment sum and apply gate, in place ----
__global__ void gate_scale(const float* __restrict__ gate,
                           const float* __restrict__ sumexp,
                           float* __restrict__ accum, int M) {
  int i = blockIdx.x * blockDim.x + threadIdx.x;
  if (i >= M * DMODEL) return;
  int node = i >> 7, c = i & (DMODEL - 1), h = c >> 4;
  float sm = sumexp[node * NHEADS + h];
  accum[i] = gate[i] * (accum[i] / (sm + 1e-12f));
}

static inline size_t align256(size_t x) { return (x + 255) & ~(size_t)255; }

extern "C" void kernel_launch(void* const* d_in, const int* in_sizes, int n_in,
                              void* d_out, int out_size, void* d_ws, size_t ws_size,
                              hipStream_t stream) {
  const float* X      = (const float*)d_in[0];
  const float* W_qkv  = (const float*)d_in[1];
  const float* w_bias = (const float*)d_in[2];
  const float* W_gate = (const float*)d_in[3];
  const float* b_gate = (const float*)d_in[4];
  const float* W_o    = (const float*)d_in[5];
  const int*   nsrc   = (const int*)d_in[6];
  const int*   ndst   = (const int*)d_in[7];

  const int M = in_sizes[0] / DMODEL;   // 50000 nodes
  const int E = in_sizes[6];            // 800000 edges

  // workspace layout
  char* p = (char*)d_ws;
  float*    qkv    = (float*)p;    p += align256((size_t)M * 3 * DMODEL * 4);
  float*    gate   = (float*)p;    p += align256((size_t)M * DMODEL * 4);
  float*    bias   = (float*)p;    p += align256((size_t)M * NHEADS * 4);
  float*    scores = (float*)p;    p += align256((size_t)E * NHEADS * 4);
  unsigned* mxbits = (unsigned*)p; p += align256((size_t)M * NHEADS * 4);
  float*    sumexp = (float*)p;    p += align256((size_t)M * NHEADS * 4);
  float*    accum  = (float*)p;    p += align256((size_t)M * DMODEL * 4);
  (void)ws_size;

  // zero the atomically-accumulated regions every call (graph-replay safe)
  hipMemsetAsync(mxbits, 0, (size_t)M * NHEADS * 4, stream);   // 0 < enc(-inf)
  hipMemsetAsync(sumexp, 0, (size_t)M * NHEADS * 4, stream);
  hipMemsetAsync(accum,  0, (size_t)M * DMODEL * 4, stream);

  const int BLK = 256;                 // 8 waves per block
  int mtiles = (M + 15) / 16;

  // 1) QKV = X @ W_qkv^T  (M x 384)
  {
    int waves = mtiles * (3 * DMODEL / 16);
    wmma_gemm_xt<0><<<(waves + 7) / 8, BLK, 0, stream>>>(X, W_qkv, nullptr, qkv,
                                                         M, 3 * DMODEL);
  }
  // 2) gate = sigmoid(X @ W_gate^T + b_gate)  (M x 128)
  {
    int waves = mtiles * (DMODEL / 16);
    wmma_gemm_xt<1><<<(waves + 7) / 8, BLK, 0, stream>>>(X, W_gate, b_gate, gate,
                                                         M, DMODEL);
  }
  // 3) bias = X @ w_bias^T  (M x 8)
  bias_kernel<<<(M * NHEADS + BLK - 1) / BLK, BLK, 0, stream>>>(X, w_bias, bias, M);

  // 4) edge scores + segment max
  edge_scores<<<(E * NHEADS + BLK - 1) / BLK, BLK, 0, stream>>>(
      qkv, bias, nsrc, ndst, scores, mxbits, E);

  // 5) exp + segment sum + weighted V aggregation
  edge_aggregate<<<(E * NHEADS + BLK - 1) / BLK, BLK, 0, stream>>>(
      qkv, scores, mxbits, nsrc, ndst, sumexp, accum, E);

  // 6) accum = gate * accum / (sum + 1e-12)
  gate_scale<<<(M * DMODEL + BLK - 1) / BLK, BLK, 0, stream>>>(gate, sumexp, accum, M);

  // 7) out = accum @ W_o^T  (M x 128)
  {
    int waves = mtiles * (DMODEL / 16);
    wmma_gemm_xt<0><<<(waves + 7) / 8, BLK, 0, stream>>>(accum, W_o, nullptr,
                                                         (float*)d_out, M, DMODEL);
  }
}